// GATInversePredictor_24137716203980
// MI455X (gfx1250) — compile-verified
//
#include <hip/hip_runtime.h>

// ---------------- problem constants ----------------
#define N_NODES   20000
#define FDIM      256          // H*C
#define HEADS     4
#define CDIM      64
#define NEDGE     320000
#define ETOT      (NEDGE + N_NODES)   // with self loops = 340000
#define NEG_SLOPE 0.2f

typedef __attribute__((ext_vector_type(2))) float v2f;
typedef __attribute__((ext_vector_type(8))) float v8f;

// ---------------- helpers ----------------
__device__ __forceinline__ unsigned f2ord(float f) {
  unsigned u = __float_as_uint(f);
  return (u & 0x80000000u) ? ~u : (u | 0x80000000u);
}
__device__ __forceinline__ float ord2f(unsigned u) {
  unsigned v = (u & 0x80000000u) ? (u ^ 0x80000000u) : ~u;
  return __uint_as_float(v);
}
__device__ __forceinline__ float leaky(float v) {
  return v > 0.0f ? v : NEG_SLOPE * v;
}

__global__ void zero_f32_kernel(float* __restrict__ p, int n) {
  int i = blockIdx.x * blockDim.x + threadIdx.x;
  if (i < n) p[i] = 0.0f;
}
__global__ void zero_u32_kernel(unsigned* __restrict__ p, int n) {
  int i = blockIdx.x * blockDim.x + threadIdx.x;
  if (i < n) p[i] = 0u;
}

// ---------------- GEMM: Out[20000x256] = A[20000x256] @ B[256x256] ----------------
// 512 threads = 16 waves. Block owns 16 rows; wave w owns 16-col tile w.
// Full-precision f32 WMMA 16x16x4 (verified: emits v_wmma_f32_16x16x4_f32).
__global__ __launch_bounds__(512)
void gemm_f32_wmma(const float* __restrict__ A, const float* __restrict__ B,
                   float* __restrict__ Out) {
  __shared__ float As[16 * 257];          // padded stride: kills LDS bank conflicts
  const int tid      = threadIdx.x;
  const int row_base = blockIdx.x * 16;

  // cooperative, coalesced load of 16x256 A tile
  for (int i = tid; i < 16 * 256; i += 512) {
    int r = i >> 8, c = i & 255;
    As[r * 257 + c] = A[(size_t)(row_base + r) * FDIM + c];
  }
  __syncthreads();

  const int wave = tid >> 5;
  const int lane = tid & 31;
  const int cn   = wave * 16;             // output column base
  const int m    = lane & 15;             // A row / B col within tile
  const int kh   = (lane >> 4) << 1;      // 0 for lanes 0-15, 2 for lanes 16-31

  v8f acc = {};
  #pragma unroll 8
  for (int k = 0; k < FDIM; k += 4) {
    v2f a, b;
    a.x = As[m * 257 + k + kh];
    a.y = As[m * 257 + k + kh + 1];
    b.x = B[(size_t)(k + kh)     * FDIM + cn + m];
    b.y = B[(size_t)(k + kh + 1) * FDIM + cn + m];
    acc = __builtin_amdgcn_wmma_f32_16x16x4_f32(false, a, false, b,
                                                (short)0, acc, false, false);
  }

  const int rbase = row_base + ((lane >> 4) << 3);
  #pragma unroll
  for (int r = 0; r < 8; ++r) {
    Out[(size_t)(rbase + r) * FDIM + cn + m] = acc[r];
  }
}

// ---------------- edge scores + segment max ----------------
// One wave per edge; lane owns 8 contiguous channels -> b128 gathers.
// Head of lane = lane>>3 (constant); per-head reduce within 8-lane groups.
__global__ __launch_bounds__(256)
void edge_scores(const float* __restrict__ xl, const float* __restrict__ xr,
                 const float* __restrict__ att, const int* __restrict__ ei,
                 float* __restrict__ score, unsigned* __restrict__ m_u) {
  const int e    = blockIdx.x * (blockDim.x >> 5) + (threadIdx.x >> 5);
  const int lane = threadIdx.x & 31;
  if (e >= ETOT) return;
  int src, dst;
  if (e < NEDGE) { src = ei[e]; dst = ei[NEDGE + e]; }
  else           { src = e - NEDGE; dst = src; }

  const float4* pl4 = (const float4*)(xl + (size_t)src * FDIM) + lane * 2;
  const float4* pr4 = (const float4*)(xr + (size_t)dst * FDIM) + lane * 2;
  const float4* at4 = (const float4*)att + lane * 2;

  float4 l0 = pl4[0], l1 = pl4[1];
  float4 r0 = pr4[0], r1 = pr4[1];
  float4 a0 = at4[0], a1 = at4[1];

  float acc = leaky(l0.x + r0.x) * a0.x + leaky(l0.y + r0.y) * a0.y +
              leaky(l0.z + r0.z) * a0.z + leaky(l0.w + r0.w) * a0.w +
              leaky(l1.x + r1.x) * a1.x + leaky(l1.y + r1.y) * a1.y +
              leaky(l1.z + r1.z) * a1.z + leaky(l1.w + r1.w) * a1.w;

  // reduce within 8-lane groups (each group = one head)
  acc += __shfl_xor(acc, 4, 32);
  acc += __shfl_xor(acc, 2, 32);
  acc += __shfl_xor(acc, 1, 32);

  if ((lane & 7) == 0) {
    const int h = lane >> 3;
    score[(size_t)e * HEADS + h] = acc;
    atomicMax(&m_u[dst * HEADS + h], f2ord(acc));
  }
}

// ---------------- p = exp(score - m[dst]); denom[dst] += p ----------------
__global__ __launch_bounds__(256)
void edge_exp(const int* __restrict__ ei, float* __restrict__ score_p,
              const unsigned* __restrict__ m_u, float* __restrict__ denom) {
  const int t = blockIdx.x * blockDim.x + threadIdx.x;
  if (t >= ETOT * HEADS) return;
  const int e = t >> 2;
  const int h = t & 3;
  const int dst = (e < NEDGE) ? ei[NEDGE + e] : (e - NEDGE);
  const float m = ord2f(m_u[dst * HEADS + h]);
  const float p = expf(score_p[t] - m);
  score_p[t] = p;
  unsafeAtomicAdd(&denom[dst * HEADS + h], p);   // native global_atomic_add_f32
}

// ---------------- weighted scatter-add aggregation ----------------
// One wave per edge; lane owns 8 contiguous channels of one head.
__global__ __launch_bounds__(256)
void edge_aggregate(const float* __restrict__ xl, const float* __restrict__ p,
                    const float* __restrict__ denom, const int* __restrict__ ei,
                    float* __restrict__ agg) {
  const int e    = blockIdx.x * (blockDim.x >> 5) + (threadIdx.x >> 5);
  const int lane = threadIdx.x & 31;
  if (e >= ETOT) return;
  int src, dst;
  if (e < NEDGE) { src = ei[e]; dst = ei[NEDGE + e]; }
  else           { src = e - NEDGE; dst = src; }

  const int h = lane >> 3;
  const float alpha = p[(size_t)e * HEADS + h] / (denom[dst * HEADS + h] + 1e-16f);

  const float4* pl4 = (const float4*)(xl + (size_t)src * FDIM) + lane * 2;
  float*        pa  = agg + (size_t)dst * FDIM + lane * 8;
  float4 v0 = pl4[0], v1 = pl4[1];

  unsafeAtomicAdd(pa + 0, alpha * v0.x);
  unsafeAtomicAdd(pa + 1, alpha * v0.y);
  unsafeAtomicAdd(pa + 2, alpha * v0.z);
  unsafeAtomicAdd(pa + 3, alpha * v0.w);
  unsafeAtomicAdd(pa + 4, alpha * v1.x);
  unsafeAtomicAdd(pa + 5, alpha * v1.y);
  unsafeAtomicAdd(pa + 6, alpha * v1.z);
  unsafeAtomicAdd(pa + 7, alpha * v1.w);
}

// ---------------- out = elu(agg + bias), in place ----------------
__global__ __launch_bounds__(256)
void finalize_elu(float* __restrict__ io, const float* __restrict__ bias, int total) {
  const int i = blockIdx.x * blockDim.x + threadIdx.x;
  if (i >= total) return;
  float v = io[i] + bias[i & (FDIM - 1)];
  io[i] = v > 0.0f ? v : (expf(v) - 1.0f);
}

// ---------------- host orchestration ----------------
extern "C" void kernel_launch(void* const* d_in, const int* in_sizes, int n_in,
                              void* d_out, int out_size, void* d_ws, size_t ws_size,
                              hipStream_t stream) {
  const float* x    = (const float*)d_in[0];
  const int*   ei   = (const int*)  d_in[1];
  const float* Wl0  = (const float*)d_in[2];
  const float* Wr0  = (const float*)d_in[3];
  const float* att0 = (const float*)d_in[4];
  const float* b0   = (const float*)d_in[5];
  const float* Wl1  = (const float*)d_in[6];
  const float* Wr1  = (const float*)d_in[7];
  const float* att1 = (const float*)d_in[8];
  const float* b1   = (const float*)d_in[9];
  float* out = (float*)d_out;

  // workspace layout (floats)
  float*    ws    = (float*)d_ws;
  float*    xl    = ws;                                   // N*256
  float*    xr    = xl   + (size_t)N_NODES * FDIM;        // N*256
  float*    hbuf  = xr   + (size_t)N_NODES * FDIM;        // N*256 (layer0 out)
  float*    pbuf  = hbuf + (size_t)N_NODES * FDIM;        // ETOT*HEADS
  unsigned* m_u   = (unsigned*)(pbuf + (size_t)ETOT * HEADS);  // N*HEADS
  float*    denom = (float*)(m_u + (size_t)N_NODES * HEADS);   // N*HEADS

  const int nh        = N_NODES * HEADS;
  const int nfeat     = N_NODES * FDIM;
  const int edgeWaves = (ETOT + 7) / 8;

  auto layer = [&](const float* in, const float* Wl, const float* Wr,
                   const float* att, const float* bias, float* outb) {
    gemm_f32_wmma<<<N_NODES / 16, 512, 0, stream>>>(in, Wl, xl);
    gemm_f32_wmma<<<N_NODES / 16, 512, 0, stream>>>(in, Wr, xr);
    zero_u32_kernel<<<(nh + 255) / 256, 256, 0, stream>>>(m_u, nh);
    zero_f32_kernel<<<(nh + 255) / 256, 256, 0, stream>>>(denom, nh);
    zero_f32_kernel<<<(nfeat + 255) / 256, 256, 0, stream>>>(outb, nfeat);
    edge_scores<<<edgeWaves, 256, 0, stream>>>(xl, xr, att, ei, pbuf, m_u);
    edge_exp<<<(ETOT * HEADS + 255) / 256, 256, 0, stream>>>(ei, pbuf, m_u, denom);
    edge_aggregate<<<edgeWaves, 256, 0, stream>>>(xl, pbuf, denom, ei, outb);
    finalize_elu<<<(nfeat + 255) / 256, 256, 0, stream>>>(outb, bias, nfeat);
  };

  layer(x,    Wl0, Wr0, att0, b0, hbuf);   // layer 0 -> hbuf
  layer(hbuf, Wl1, Wr1, att1, b1, out);    // layer 1 -> d_out
}